// SurvivalGNN_2302102471069
// MI455X (gfx1250) — compile-verified
//
#include <hip/hip_runtime.h>
#include <hip/hip_bf16.h>

typedef float v2f __attribute__((ext_vector_type(2)));
typedef float v8f __attribute__((ext_vector_type(8)));

#define NDIM 64   // feature width for both GCN layers
#define HIDW 128  // final MLP hidden

// ---------------- utility: zero / init ----------------
__global__ void k_zero_f32(float* __restrict__ p, int n) {
    int i = blockIdx.x * blockDim.x + threadIdx.x;
    if (i < n) p[i] = 0.0f;
}

__global__ void k_init_one(float* __restrict__ p, int n) {
    int i = blockIdx.x * blockDim.x + threadIdx.x;
    if (i < n) p[i] = 1.0f;  // self-loop contributes 1 to degree
}

// ---------------- degree accumulation over dst ----------------
__global__ void k_deg(const int* __restrict__ dst, float* __restrict__ deg, int ne) {
    int e = blockIdx.x * blockDim.x + threadIdx.x;
    if (e < ne) atomicAdd(&deg[dst[e]], 1.0f);
}

__global__ void k_rsqrt_inplace(float* __restrict__ deg, int n) {
    int i = blockIdx.x * blockDim.x + threadIdx.x;
    if (i < n) deg[i] = rsqrtf(deg[i]);
}

// ---------------- WMMA GEMM: O[N x 64] = A[N x 64] @ W[64 x 64] ----------------
// block = 128 threads (4 waves). Wave w computes the 16x16 tile at
// rows [16*blockIdx.x, +16), cols [16*w, +16).  K = 64 = 16 steps of K=4.
// f32 WMMA keeps full fp32 precision.
__global__ void k_gemm64_wmma(const float* __restrict__ A,
                              const float* __restrict__ W,
                              float* __restrict__ O) {
    const int wave = threadIdx.x >> 5;
    const int lane = threadIdx.x & 31;
    const int half = lane >> 4;       // 0: K lo pair, 1: K hi pair
    const int l16  = lane & 15;
    const int rb = blockIdx.x * 16;   // row base
    const int cb = wave * 16;         // col base

    const float* __restrict__ arow = A + (size_t)(rb + l16) * NDIM;

    v8f acc = {};
#pragma unroll
    for (int kc = 0; kc < 16; ++kc) {
        const int k0 = kc * 4 + half * 2;
        // A 16x4 tile: lanes 0-15 -> K={0,1}, lanes 16-31 -> K={2,3}
        v2f a;
        a.x = arow[k0];
        a.y = arow[k0 + 1];
        // B 4x16 tile (mirror layout, N across lanes)
        v2f b;
        b.x = W[(size_t)(k0)     * NDIM + cb + l16];
        b.y = W[(size_t)(k0 + 1) * NDIM + cb + l16];
        acc = __builtin_amdgcn_wmma_f32_16x16x4_f32(
            /*neg_a=*/false, a, /*neg_b=*/false, b,
            /*c_mod=*/(short)0, acc, /*reuse_a=*/false, /*reuse_b=*/false);
    }
    // C/D layout: VGPR r -> M=r (lanes 0-15), M=r+8 (lanes 16-31); N = l16
#pragma unroll
    for (int r = 0; r < 8; ++r) {
        const int m = r + half * 8;
        O[(size_t)(rb + m) * NDIM + cb + l16] = acc[r];
    }
}

// ---------------- edge scatter: ACC[d] += H[s] * dinv[s]*dinv[d] ----------------
// one wave32 per edge; each lane moves a float2 (256B coalesced gather/edge)
__global__ void k_edge_scatter(const int* __restrict__ src,
                               const int* __restrict__ dst,
                               const float* __restrict__ dinv,
                               const float* __restrict__ H,
                               float* __restrict__ ACC, int ne) {
    const int wavesPerBlk = blockDim.x >> 5;
    const int e = blockIdx.x * wavesPerBlk + (threadIdx.x >> 5);
    if (e >= ne) return;
    const int lane = threadIdx.x & 31;
    const int s = src[e];
    const int d = dst[e];
    const float norm = dinv[s] * dinv[d];
    const float2 v = ((const float2*)(H + (size_t)s * NDIM))[lane];
    float* o = ACC + (size_t)d * NDIM + lane * 2;
    atomicAdd(o,     v.x * norm);
    atomicAdd(o + 1, v.y * norm);
}

// ---------------- self-loop + bias + ReLU ----------------
// OUT[v,f] = relu(ACC[v,f] + H[v,f]*dinv[v]^2 + b[f])
__global__ void k_selfloop_bias_relu(const float* __restrict__ ACC,
                                     const float* __restrict__ H,
                                     const float* __restrict__ dinv,
                                     const float* __restrict__ bias,
                                     float* __restrict__ OUT, int total) {
    int i = blockIdx.x * blockDim.x + threadIdx.x;
    if (i >= total) return;
    const int v = i >> 6;       // /64
    const int f = i & 63;
    const float di = dinv[v];
    const float x = ACC[i] + H[i] * di * di + bias[f];
    OUT[i] = fmaxf(x, 0.0f);
}

// ---------------- pooling: sums per graph + counts ----------------
__global__ void k_pool(const float* __restrict__ HC,
                       const int* __restrict__ batch_ids,
                       float* __restrict__ pool,   // [B,64]
                       float* __restrict__ cnt,    // [B]
                       int total) {
    int i = blockIdx.x * blockDim.x + threadIdx.x;
    if (i >= total) return;
    const int v = i >> 6;
    const int f = i & 63;
    const int b = batch_ids[v];
    atomicAdd(&pool[b * NDIM + f], HC[i]);
    if (f == 0) atomicAdd(&cnt[b], 1.0f);
}

// ---------------- final MLP: out[b] = (relu(G@Wf1+bf1))@Wf2 + bf2 ----------------
// grid = B blocks, block = 128 threads. G[b,k] = pool[b,k]/max(cnt[b],1)
__global__ void k_mlp(const float* __restrict__ pool,
                      const float* __restrict__ cnt,
                      const float* __restrict__ Wf1,  // [64,128]
                      const float* __restrict__ bf1,  // [128]
                      const float* __restrict__ Wf2,  // [128,1]
                      const float* __restrict__ bf2,  // [1]
                      float* __restrict__ out) {      // [B]
    __shared__ float red[HIDW];
    const int b = blockIdx.x;
    const int j = threadIdx.x;
    const float inv = 1.0f / fmaxf(cnt[b], 1.0f);
    float acc = bf1[j];
#pragma unroll
    for (int k = 0; k < NDIM; ++k)
        acc = fmaf(pool[b * NDIM + k] * inv, Wf1[k * HIDW + j], acc);
    const float h = fmaxf(acc, 0.0f);
    red[j] = h * Wf2[j];
    __syncthreads();
    for (int s = HIDW / 2; s > 0; s >>= 1) {
        if (j < s) red[j] += red[j + s];
        __syncthreads();
    }
    if (j == 0) out[b] = red[0] + bf2[0];
}

extern "C" void kernel_launch(void* const* d_in, const int* in_sizes, int n_in,
                              void* d_out, int out_size, void* d_ws, size_t ws_size,
                              hipStream_t stream) {
    // inputs: V, E, batch_ids, W1, b1, W2, b2, Wf1, bf1, Wf2, bf2
    const float* V   = (const float*)d_in[0];
    const int*   E   = (const int*)  d_in[1];
    const int*   bid = (const int*)  d_in[2];
    const float* W1  = (const float*)d_in[3];
    const float* b1  = (const float*)d_in[4];
    const float* W2  = (const float*)d_in[5];
    const float* b2  = (const float*)d_in[6];
    const float* Wf1 = (const float*)d_in[7];
    const float* bf1 = (const float*)d_in[8];
    const float* Wf2 = (const float*)d_in[9];
    const float* bf2 = (const float*)d_in[10];
    float* out = (float*)d_out;

    const int N  = in_sizes[0] / NDIM;   // 100000
    const int NE = in_sizes[1] / 2;      // 1600000
    const int B  = out_size;             // 64
    const int total = N * NDIM;

    const int* src = E;
    const int* dst = E + NE;

    // workspace layout (floats)
    float* ws   = (float*)d_ws;
    float* H    = ws;                          // [N,64] linear transform result
    float* ACC  = H   + (size_t)N * NDIM;      // [N,64] scatter accumulator
    float* HC   = ACC + (size_t)N * NDIM;      // [N,64] activated output
    float* dinv = HC  + (size_t)N * NDIM;      // [N]    degree -> rsqrt(degree)
    float* pool = dinv + N;                    // [B,64]
    float* cnt  = pool + (size_t)B * NDIM;     // [B]

    const int TB = 256;
    dim3 blk(TB);

    // 1) degrees (self-loop = 1), then dinv = rsqrt(deg)
    k_init_one<<<(N + TB - 1) / TB, blk, 0, stream>>>(dinv, N);
    k_deg<<<(NE + TB - 1) / TB, blk, 0, stream>>>(dst, dinv, NE);
    k_rsqrt_inplace<<<(N + TB - 1) / TB, blk, 0, stream>>>(dinv, N);

    const int gemmBlocks = N / 16;                 // N divisible by 16
    const int edgeBlocks = (NE + 7) / 8;           // 8 waves/block, 1 edge/wave
    const int elemBlocks = (total + TB - 1) / TB;

    // ---- layer 1 ----
    k_zero_f32<<<elemBlocks, blk, 0, stream>>>(ACC, total);
    k_gemm64_wmma<<<gemmBlocks, dim3(128), 0, stream>>>(V, W1, H);
    k_edge_scatter<<<edgeBlocks, blk, 0, stream>>>(src, dst, dinv, H, ACC, NE);
    k_selfloop_bias_relu<<<elemBlocks, blk, 0, stream>>>(ACC, H, dinv, b1, HC, total);

    // ---- layer 2 ----
    k_zero_f32<<<elemBlocks, blk, 0, stream>>>(ACC, total);
    k_gemm64_wmma<<<gemmBlocks, dim3(128), 0, stream>>>(HC, W2, H);
    k_edge_scatter<<<edgeBlocks, blk, 0, stream>>>(src, dst, dinv, H, ACC, NE);
    k_selfloop_bias_relu<<<elemBlocks, blk, 0, stream>>>(ACC, H, dinv, b2, HC, total);

    // ---- global mean pool + MLP head ----
    k_zero_f32<<<(B * NDIM + B + TB - 1) / TB, blk, 0, stream>>>(pool, B * NDIM + B);
    k_pool<<<elemBlocks, blk, 0, stream>>>(HC, bid, pool, cnt, total);
    k_mlp<<<B, dim3(HIDW), 0, stream>>>(pool, cnt, Wf1, bf1, Wf2, bf2, out);
}